// CTDExplainer_49520972923523
// MI455X (gfx1250) — compile-verified
//
#include <hip/hip_runtime.h>

typedef __attribute__((ext_vector_type(2))) float v2f;
typedef __attribute__((ext_vector_type(8))) float v8f;

#define N_X   6144
#define N_EMB 2048
#define NTOT  8192
#define DD    64
#define NE    262144

__device__ __forceinline__ v8f wmma_f32(v2f a, v2f b, v8f c) {
  // V_WMMA_F32_16X16X4_F32 : D = A(16x4) x B(4x16) + C(16x16), exact fp32
  return __builtin_amdgcn_wmma_f32_16x16x4_f32(
      /*neg_a=*/false, a, /*neg_b=*/false, b,
      /*c_mod=*/(short)0, c, /*reuse_a=*/false, /*reuse_b=*/false);
}

// ---------------- build H = concat(x, emb) ----------------
__global__ void build_H_kernel(const float* __restrict__ x,
                               const float* __restrict__ emb,
                               float* __restrict__ H) {
  int i = blockIdx.x * blockDim.x + threadIdx.x;
  if (i >= NTOT * DD) return;
  H[i] = (i < N_X * DD) ? x[i] : emb[i - N_X * DD];
}

// ---------------- small GEMM: out[N,64] = phi(A[N,64]) @ W[64,64] ----------------
// One wave -> 16 rows x 64 cols, K=64 via 16 x (16x16x4) WMMA steps per 16-col tile.
// RELU is a compile-time flag so no cndmask select stream is generated.
template <int RELU>
__global__ __launch_bounds__(256) void gemm_AW_kernel(
    const float* __restrict__ A, const float* __restrict__ W,
    float* __restrict__ out) {
  int wave = (blockIdx.x * blockDim.x + threadIdx.x) >> 5;
  int lane = threadIdx.x & 31;
  int m0 = wave * 16;
  if (m0 >= NTOT) return;

  int khalf = (lane >> 4) * 2;            // lanes 16-31 hold K+2 (ISA A layout)
  const float* arow = A + (size_t)(m0 + (lane & 15)) * DD;

  v2f afrag[16];
#pragma unroll
  for (int s = 0; s < 16; ++s) {
    int k = 4 * s + khalf;
    float a0 = arow[k], a1 = arow[k + 1];
    if (RELU) { a0 = fmaxf(a0, 0.f); a1 = fmaxf(a1, 0.f); }
    afrag[s] = {a0, a1};
  }

  int ncol = lane & 15;
#pragma unroll
  for (int t = 0; t < 4; ++t) {
    int nact = t * 16 + ncol;
    v8f acc = {};
#pragma unroll
    for (int s = 0; s < 16; ++s) {
      int k = 4 * s + khalf;
      v2f bfrag = {W[k * DD + nact], W[(k + 1) * DD + nact]};
      acc = wmma_f32(afrag[s], bfrag, acc);
    }
    int mbase = m0 + (lane >> 4) * 8;     // C/D layout: M = v + 8*(lane/16)
#pragma unroll
    for (int v = 0; v < 8; ++v)
      out[(size_t)(mbase + v) * DD + nact] = acc[v];
  }
}

// ---------------- big GEMM: z[N,N] = A[N,64] @ B[N,64]^T ----------------
// One wave -> 16(M) x 64(N) strip; A-fragments held in regs, reused over 4 B tiles.
// z stores are non-temporal: 268MB/layer never fits L2; keep L2 for H/HB/HW.
__global__ __launch_bounds__(256) void gemm_ABT_kernel(
    const float* __restrict__ A, const float* __restrict__ B,
    float* __restrict__ z) {
  int wave = (blockIdx.x * blockDim.x + threadIdx.x) >> 5;
  int lane = threadIdx.x & 31;
  const int TN = NTOT / 64;               // 128 col-strips
  int tm = wave / TN;
  int tn = wave - tm * TN;
  if (tm >= NTOT / 16) return;
  int m0 = tm * 16;
  int j0 = tn * 64;

  int khalf = (lane >> 4) * 2;
  const float* arow = A + (size_t)(m0 + (lane & 15)) * DD;
  v2f afrag[16];
#pragma unroll
  for (int s = 0; s < 16; ++s) {
    int k = 4 * s + khalf;
    afrag[s] = {arow[k], arow[k + 1]};
  }

#pragma unroll
  for (int t = 0; t < 4; ++t) {
    // B^T tile: lane n holds row H[j0+16t+n][k..k+1] -> exactly B-fragment layout
    const float* brow = B + (size_t)(j0 + t * 16 + (lane & 15)) * DD;
    v8f acc = {};
#pragma unroll
    for (int s = 0; s < 16; ++s) {
      int k = 4 * s + khalf;
      v2f bfrag = {brow[k], brow[k + 1]};
      acc = wmma_f32(afrag[s], bfrag, acc);
    }
    int mbase = m0 + (lane >> 4) * 8;
    int col = j0 + t * 16 + (lane & 15);
#pragma unroll
    for (int v = 0; v < 8; ++v)
      __builtin_nontemporal_store(acc[v], &z[(size_t)(mbase + v) * NTOT + col]);
  }
}

// ---------------- a_e = z[row,col] * edge_val ----------------
__global__ void edge_gather_kernel(const float* __restrict__ z,
                                   const int* __restrict__ erow,
                                   const int* __restrict__ ecol,
                                   const float* __restrict__ eval,
                                   float* __restrict__ a_e) {
  int e = blockIdx.x * blockDim.x + threadIdx.x;
  if (e >= NE) return;
  float zv = __builtin_nontemporal_load(&z[(size_t)erow[e] * NTOT + ecol[e]]);
  a_e[e] = zv * eval[e];
}

__global__ void zero_f32_kernel(float* __restrict__ p, int n) {
  int i = blockIdx.x * blockDim.x + threadIdx.x;
  if (i < n) p[i] = 0.f;
}

// ---------------- agg[row] += HW[etype][col] * a_e ----------------
// 32 lanes per edge, 2 floats/lane via float2 message loads.
__global__ void edge_scatter_kernel(const float* __restrict__ HW,  // [3,N,64]
                                    const int* __restrict__ erow,
                                    const int* __restrict__ ecol,
                                    const int* __restrict__ etyp,
                                    const float* __restrict__ a_e,
                                    float* __restrict__ agg) {
  long tid = (long)blockIdx.x * blockDim.x + threadIdx.x;
  int e = (int)(tid >> 5);
  int d = (int)(tid & 31) * 2;
  if (e >= NE) return;
  float av = a_e[e];
  const float2* src = (const float2*)&HW[(((size_t)etyp[e] * NTOT + ecol[e]) * DD) + d];
  float2 m = *src;
  float* dst = &agg[(size_t)erow[e] * DD + d];
  atomicAdd(dst,     m.x * av);
  atomicAdd(dst + 1, m.y * av);
}

// ---------------- out = LN?(agg + hroot + bias) ----------------
// one wave per row, 2 elems/lane, wave32 shuffle reduction for mean/var
__global__ __launch_bounds__(256) void combine_ln_kernel(
    const float* __restrict__ agg, const float* __restrict__ hroot,
    const float* __restrict__ bias, const float* __restrict__ g,
    const float* __restrict__ b, float* __restrict__ out, int do_ln) {
  int wave = (blockIdx.x * blockDim.x + threadIdx.x) >> 5;
  int lane = threadIdx.x & 31;
  if (wave >= NTOT) return;
  int base = wave * DD + lane * 2;
  float h0 = agg[base]     + hroot[base]     + bias[lane * 2];
  float h1 = agg[base + 1] + hroot[base + 1] + bias[lane * 2 + 1];
  if (do_ln) {
    float s  = h0 + h1;
    float s2 = h0 * h0 + h1 * h1;
#pragma unroll
    for (int off = 16; off >= 1; off >>= 1) {
      s  += __shfl_xor(s,  off, 32);
      s2 += __shfl_xor(s2, off, 32);
    }
    float mu  = s * (1.0f / 64.0f);
    float var = s2 * (1.0f / 64.0f) - mu * mu;
    float r   = rsqrtf(var + 1e-5f);
    h0 = (h0 - mu) * r * g[lane * 2]     + b[lane * 2];
    h1 = (h1 - mu) * r * g[lane * 2 + 1] + b[lane * 2 + 1];
  }
  out[base] = h0;
  out[base + 1] = h1;
}

extern "C" void kernel_launch(void* const* d_in, const int* in_sizes, int n_in,
                              void* d_out, int out_size, void* d_ws, size_t ws_size,
                              hipStream_t stream) {
  const float* x     = (const float*)d_in[0];
  const float* emb   = (const float*)d_in[1];
  const float* bilin = (const float*)d_in[2];   // [3,64,64]
  const float* eval  = (const float*)d_in[3];   // [E]
  const float* Wroot = (const float*)d_in[4];   // [3,64,64]
  const float* Wrel  = (const float*)d_in[5];   // [3,3,64,64]
  const float* bias  = (const float*)d_in[6];   // [3,64]
  const float* lng   = (const float*)d_in[7];   // [2,64]
  const float* lnb   = (const float*)d_in[8];   // [2,64]
  const int* erow = (const int*)d_in[9];
  const int* ecol = (const int*)d_in[10];
  const int* etyp = (const int*)d_in[11];

  float* out  = (float*)d_out;
  float* outH = out;                            // [N,64]  (tuple elem 0)
  float* outZ = out + (size_t)NTOT * DD;        // [3,N,N] (tuple elem 1)

  float* ws    = (float*)d_ws;
  float* H     = ws;                            // N*64
  float* HB    = H     + (size_t)NTOT * DD;     // N*64
  float* HW    = HB    + (size_t)NTOT * DD;     // 3*N*64
  float* Hroot = HW    + (size_t)3 * NTOT * DD; // N*64
  float* agg   = Hroot + (size_t)NTOT * DD;     // N*64
  float* a_e   = agg   + (size_t)NTOT * DD;     // E

  build_H_kernel<<<(NTOT * DD) / 256, 256, 0, stream>>>(x, emb, H);

  const int smallGemmBlocks = (NTOT / 16) * 32 / 256;                 // 64
  const int bigGemmBlocks   = (NTOT / 16) * (NTOT / 64) * 32 / 256;   // 8192

  for (int l = 0; l < 3; ++l) {
    float* z = outZ + (size_t)l * NTOT * NTOT;

    // HB = H @ bilin[l]   (z always uses non-ReLU'd H)
    gemm_AW_kernel<0><<<smallGemmBlocks, 256, 0, stream>>>(
        H, bilin + (size_t)l * DD * DD, HB);

    // z_l = HB @ H^T  -> straight into d_out (store-bandwidth bound, NT stores)
    gemm_ABT_kernel<<<bigGemmBlocks, 256, 0, stream>>>(HB, H, z);

    // a_e = z[row,col] * edge_val
    edge_gather_kernel<<<NE / 256, 256, 0, stream>>>(z, erow, ecol, eval, a_e);

    // per-relation transforms (+ root), ReLU on input for layers 1,2
    if (l == 0) {
      for (int r = 0; r < 3; ++r)
        gemm_AW_kernel<0><<<smallGemmBlocks, 256, 0, stream>>>(
            H, Wrel + ((size_t)l * 3 + r) * DD * DD, HW + (size_t)r * NTOT * DD);
      gemm_AW_kernel<0><<<smallGemmBlocks, 256, 0, stream>>>(
          H, Wroot + (size_t)l * DD * DD, Hroot);
    } else {
      for (int r = 0; r < 3; ++r)
        gemm_AW_kernel<1><<<smallGemmBlocks, 256, 0, stream>>>(
            H, Wrel + ((size_t)l * 3 + r) * DD * DD, HW + (size_t)r * NTOT * DD);
      gemm_AW_kernel<1><<<smallGemmBlocks, 256, 0, stream>>>(
          H, Wroot + (size_t)l * DD * DD, Hroot);
    }

    zero_f32_kernel<<<(NTOT * DD) / 256, 256, 0, stream>>>(agg, NTOT * DD);

    // agg[row] += HW[etype][col] * a_e   (32 lanes/edge, float2 messages)
    edge_scatter_kernel<<<((long)NE * 32) / 256, 256, 0, stream>>>(
        HW, erow, ecol, etyp, a_e, agg);

    // H = LN(agg + Hroot + bias)  (no LN on final layer; final goes to d_out)
    float* dst = (l == 2) ? outH : H;
    int gidx = (l < 2) ? l : 0;  // ln params unused when do_ln==0
    combine_ln_kernel<<<NTOT * 32 / 256, 256, 0, stream>>>(
        agg, Hroot, bias + (size_t)l * DD,
        lng + (size_t)gidx * DD, lnb + (size_t)gidx * DD, dst, (l < 2) ? 1 : 0);
  }
}